// StructurePredictor_68564857913753
// MI455X (gfx1250) — compile-verified
//
#include <hip/hip_runtime.h>
#include <hip/hip_bf16.h>

// ---------------------------------------------------------------------------
// StructurePredictor on MI455X (gfx1250, wave32).
// fp32 end-to-end; matrix work on V_WMMA_F32_16X16X4_F32.
// Each wave owns a 32x32 C tile = 2x2 register-blocked 16x16 WMMA tiles.
// ---------------------------------------------------------------------------

typedef __attribute__((ext_vector_type(2))) float v2f;
typedef __attribute__((ext_vector_type(8))) float v8f;

#define MROWS 512   // B*L
#define HDIM  320
#define SEQ_B 2
#define SEQ_L 256
#define NHEAD 8
#define HEADD 40

#define WMMA4(A_, B_, C_) \
    __builtin_amdgcn_wmma_f32_16x16x4_f32(false, (A_), false, (B_), (short)0, (C_), false, false)

// ---------------------------------------------------------------------------
// fp32 WMMA GEMM: C[M x N] = act(A[M x K] (ld=lda) @ W[K x N] (ld=N) + bias)
// M, N multiples of 32; K arbitrary (branchless masked ragged tail).
// All control flow is wave-uniform -> EXEC all-ones at every WMMA.
// ---------------------------------------------------------------------------
template <bool RELU>
__global__ void wmma_gemm_f32(const float* __restrict__ A,
                              const float* __restrict__ W,
                              const float* __restrict__ bias,
                              float* __restrict__ C,
                              int M, int N, int K, int lda) {
    const int lane   = threadIdx.x & 31;
    const int wave   = threadIdx.x >> 5;
    const int tilesN = N >> 5;               // 32-wide tiles
    const int total  = (M >> 5) * tilesN;
    const int tile   = blockIdx.x * (blockDim.x >> 5) + wave;
    if (tile >= total) return;               // wave-uniform exit

    const int tm   = (tile / tilesN) << 5;
    const int tn   = (tile % tilesN) << 5;
    const int half = lane >> 4;              // lanes 0..15 -> K+{0,1}; 16..31 -> K+{2,3}
    const int l16  = lane & 15;

    const float* __restrict__ Ar0 = A + (size_t)(tm + l16) * lda;       // rows tm..tm+15
    const float* __restrict__ Ar1 = Ar0 + (size_t)16 * lda;             // rows tm+16..tm+31
    const float* __restrict__ Wc0 = W + tn + l16;                       // cols tn..tn+15
    const float* __restrict__ Wc1 = Wc0 + 16;                           // cols tn+16..tn+31

    v8f acc00 = {}, acc01 = {}, acc10 = {}, acc11 = {};

    const int Kmain = K & ~3;
    int ka = 2 * half;
    #pragma unroll 2
    for (int k0 = 0; k0 < Kmain; k0 += 4, ka += 4) {
        v2f a0, a1, b0, b1;
        a0.x = Ar0[ka];  a0.y = Ar0[ka + 1];
        a1.x = Ar1[ka];  a1.y = Ar1[ka + 1];
        const size_t w0 = (size_t)ka * N, w1 = (size_t)(ka + 1) * N;
        b0.x = Wc0[w0];  b0.y = Wc0[w1];
        b1.x = Wc1[w0];  b1.y = Wc1[w1];
        acc00 = WMMA4(a0, b0, acc00);
        acc01 = WMMA4(a0, b1, acc01);
        acc10 = WMMA4(a1, b0, acc10);
        acc11 = WMMA4(a1, b1, acc11);
    }

    if (Kmain < K) {
        // Ragged tail (K=323, K=7): unconditional loads from clamped (valid)
        // addresses, zeroed via 0/1 mask multiply. No divergent branches.
        const int ka0 = Kmain + 2 * half;
        const int ka1 = ka0 + 1;
        const int c0  = (ka0 < K) ? ka0 : (K - 1);
        const int c1  = (ka1 < K) ? ka1 : (K - 1);
        const float m0 = (ka0 < K) ? 1.0f : 0.0f;
        const float m1 = (ka1 < K) ? 1.0f : 0.0f;
        v2f a0, a1, b0, b1;
        a0.x = Ar0[c0] * m0;  a0.y = Ar0[c1] * m1;
        a1.x = Ar1[c0] * m0;  a1.y = Ar1[c1] * m1;
        const size_t w0 = (size_t)c0 * N, w1 = (size_t)c1 * N;
        b0.x = Wc0[w0] * m0;  b0.y = Wc0[w1] * m1;
        b1.x = Wc1[w0] * m0;  b1.y = Wc1[w1] * m1;
        acc00 = WMMA4(a0, b0, acc00);
        acc01 = WMMA4(a0, b1, acc01);
        acc10 = WMMA4(a1, b0, acc10);
        acc11 = WMMA4(a1, b1, acc11);
    }

    // C/D layout: VGPR r -> row (+r + 8*half), col l16 within each 16x16 tile.
    const int col0 = tn + l16;
    const int col1 = col0 + 16;
    float bias0 = 0.0f, bias1 = 0.0f;
    if (bias) { bias0 = bias[col0]; bias1 = bias[col1]; }
    #pragma unroll
    for (int r = 0; r < 8; ++r) {
        const int row0 = tm + r + 8 * half;
        const int row1 = row0 + 16;
        float v00 = acc00[r] + bias0;
        float v01 = acc01[r] + bias1;
        float v10 = acc10[r] + bias0;
        float v11 = acc11[r] + bias1;
        if (RELU) {
            v00 = fmaxf(v00, 0.0f); v01 = fmaxf(v01, 0.0f);
            v10 = fmaxf(v10, 0.0f); v11 = fmaxf(v11, 0.0f);
        }
        C[(size_t)row0 * N + col0] = v00;
        C[(size_t)row0 * N + col1] = v01;
        C[(size_t)row1 * N + col0] = v10;
        C[(size_t)row1 * N + col1] = v11;
    }
}

// ---------------------------------------------------------------------------
// Tiny-N projection (N = 3, 4, 7): one thread per output element.
// ---------------------------------------------------------------------------
template <bool RELU>
__global__ void small_gemm_f32(const float* __restrict__ A,
                               const float* __restrict__ W,
                               const float* __restrict__ bias,
                               float* __restrict__ C,
                               int M, int N, int K, int lda) {
    const int idx = blockIdx.x * blockDim.x + threadIdx.x;
    if (idx >= M * N) return;
    const int r = idx / N, n = idx % N;
    const float* __restrict__ a = A + (size_t)r * lda;
    float s = bias[n];
    for (int k = 0; k < K; ++k) s += a[k] * W[(size_t)k * N + n];
    if (RELU) s = fmaxf(s, 0.0f);
    C[idx] = s;
}

// ---------------------------------------------------------------------------
// sc_in = concat([x (512x320), backbone (512x3)], -1) -> 512x323
// ---------------------------------------------------------------------------
__global__ void build_scin(const float* __restrict__ x,
                           const float* __restrict__ bb,
                           float* __restrict__ scin) {
    const int idx = blockIdx.x * blockDim.x + threadIdx.x;
    if (idx >= MROWS * 323) return;
    const int r = idx / 323, c = idx % 323;
    scin[idx] = (c < HDIM) ? x[(size_t)r * HDIM + c] : bb[r * 3 + (c - HDIM)];
}

// comb = concat([backbone (512x3), side_chain (512x4)], -1) -> 512x7
__global__ void build_comb(const float* __restrict__ bb,
                           const float* __restrict__ sc,
                           float* __restrict__ comb) {
    const int idx = blockIdx.x * blockDim.x + threadIdx.x;
    if (idx >= MROWS * 7) return;
    const int r = idx / 7, c = idx % 7;
    comb[idx] = (c < 3) ? bb[r * 3 + c] : sc[r * 4 + (c - 3)];
}

// ---------------------------------------------------------------------------
// MHA with sequence length S=2 (softmax over 2 logits, closed form).
// qkv: 512 x 960 (row = s*256 + n ; q|k|v sections of 320, head h at h*40).
// One thread per (s, n, head).
// ---------------------------------------------------------------------------
__global__ void attention2_kernel(const float* __restrict__ qkv,
                                  float* __restrict__ out) {
    const int idx = blockIdx.x * blockDim.x + threadIdx.x;
    if (idx >= SEQ_B * SEQ_L * NHEAD) return;
    const int h = idx & (NHEAD - 1);
    const int n = (idx >> 3) & (SEQ_L - 1);
    const int s = idx >> 11;

    const float scale = 0.15811388300841897f;  // 1/sqrt(40)
    const float* __restrict__ q  = qkv + (size_t)(s * SEQ_L + n) * 960 + h * HEADD;
    const float* __restrict__ k0 = qkv + (size_t)(0 * SEQ_L + n) * 960 + HDIM + h * HEADD;
    const float* __restrict__ k1 = qkv + (size_t)(1 * SEQ_L + n) * 960 + HDIM + h * HEADD;

    float s0 = 0.0f, s1 = 0.0f;
    #pragma unroll 8
    for (int d = 0; d < HEADD; ++d) {
        s0 += q[d] * k0[d];
        s1 += q[d] * k1[d];
    }
    s0 *= scale; s1 *= scale;
    const float m  = fmaxf(s0, s1);
    float e0 = __expf(s0 - m), e1 = __expf(s1 - m);
    const float inv = 1.0f / (e0 + e1);
    e0 *= inv; e1 *= inv;

    const float* __restrict__ v0 = qkv + (size_t)(0 * SEQ_L + n) * 960 + 2 * HDIM + h * HEADD;
    const float* __restrict__ v1 = qkv + (size_t)(1 * SEQ_L + n) * 960 + 2 * HDIM + h * HEADD;
    float* __restrict__ o = out + (size_t)(s * SEQ_L + n) * HDIM + h * HEADD;
    #pragma unroll 8
    for (int d = 0; d < HEADD; ++d) o[d] = e0 * v0[d] + e1 * v1[d];
}

// ---------------------------------------------------------------------------
// contact_map[b,i,j] = sum_c relu(hi[b,i,c] + hj[b,j,c] + b1[c]) * w2[c] + b2
// Block: 16x16 (i,j) tile; hi/hj rows + b1/w2 staged in LDS.
// LDS stride padded to 164 (164 % 64 banks = 36, gcd 4 -> 16 distinct banks).
// ---------------------------------------------------------------------------
#define CDIM 160
#define CPAD 164
__global__ void contact_kernel(const float* __restrict__ hi,
                               const float* __restrict__ hj,
                               const float* __restrict__ b1,
                               const float* __restrict__ w2,
                               const float* __restrict__ b2p,
                               float* __restrict__ out) {
    __shared__ float shi[16][CPAD];
    __shared__ float shj[16][CPAD];
    __shared__ float sb[CDIM];
    __shared__ float sw[CDIM];

    const int b  = blockIdx.z;
    const int i0 = blockIdx.y << 4;
    const int j0 = blockIdx.x << 4;
    const int tid = threadIdx.y * 16 + threadIdx.x;

    for (int t = tid; t < 16 * CDIM; t += 256) {
        const int r = t / CDIM, c = t % CDIM;
        shi[r][c] = hi[(size_t)(b * SEQ_L + i0 + r) * CDIM + c];
        shj[r][c] = hj[(size_t)(b * SEQ_L + j0 + r) * CDIM + c];
    }
    if (tid < CDIM) { sb[tid] = b1[tid]; sw[tid] = w2[tid]; }
    __syncthreads();

    const int i = threadIdx.y, j = threadIdx.x;
    float acc = b2p[0];
    #pragma unroll 8
    for (int c = 0; c < CDIM; ++c)
        acc += fmaxf(shi[i][c] + shj[j][c] + sb[c], 0.0f) * sw[c];

    out[(size_t)(b * SEQ_L + i0 + i) * SEQ_L + j0 + j] = acc;
}

// ---------------------------------------------------------------------------
// Host-side launch
// ---------------------------------------------------------------------------
extern "C" void kernel_launch(void* const* d_in, const int* in_sizes, int n_in,
                              void* d_out, int out_size, void* d_ws, size_t ws_size,
                              hipStream_t stream) {
    const float* x         = (const float*)d_in[0];   // (2,256,320)
    const float* in_proj_w = (const float*)d_in[1];   // (320,960)
    const float* in_proj_b = (const float*)d_in[2];   // (960)
    const float* out_proj_w= (const float*)d_in[3];   // (320,320)
    const float* out_proj_b= (const float*)d_in[4];   // (320)
    const float* mlp_w1    = (const float*)d_in[5];   // (640,160)
    const float* mlp_b1    = (const float*)d_in[6];   // (160)
    const float* mlp_w2    = (const float*)d_in[7];   // (160,1)
    const float* mlp_b2    = (const float*)d_in[8];   // (1)
    const float* bb_w1     = (const float*)d_in[9];   // (320,640)
    const float* bb_b1     = (const float*)d_in[10];  // (640)
    const float* bb_w2     = (const float*)d_in[11];  // (640,3)
    const float* bb_b2     = (const float*)d_in[12];  // (3)
    const float* sc_w1     = (const float*)d_in[13];  // (323,320)
    const float* sc_b1     = (const float*)d_in[14];  // (320)
    const float* sc_w2     = (const float*)d_in[15];  // (320,160)
    const float* sc_b2     = (const float*)d_in[16];  // (160)
    const float* sc_w3     = (const float*)d_in[17];  // (160,4)
    const float* sc_b3     = (const float*)d_in[18];  // (4)
    const float* ref_w1    = (const float*)d_in[19];  // (7,128)
    const float* ref_b1    = (const float*)d_in[20];  // (128)
    const float* ref_w2    = (const float*)d_in[21];  // (128,256)
    const float* ref_b2    = (const float*)d_in[22];  // (256)
    const float* ref_w3    = (const float*)d_in[23];  // (256,7)
    const float* ref_b3    = (const float*)d_in[24];  // (7)

    float* out     = (float*)d_out;
    float* bb_out  = out;                  // 512*3   = 1536
    float* sc_out  = out + 1536;           // 512*4   = 2048
    float* cm_out  = out + 3584;           // 2*256^2 = 131072
    float* rf_out  = out + 134656;         // 512*7   = 3584

    float* w = (float*)d_ws;
    float* t1       = w;                   // 512*640
    float* scin     = t1       + 512*640;  // 512*323
    float* h1       = scin     + 512*323;  // 512*320
    float* h2       = h1       + 512*320;  // 512*160
    float* qkv      = h2       + 512*160;  // 512*960
    float* attn_out = qkv      + 512*960;  // 512*320
    float* a_proj   = attn_out + 512*320;  // 512*320
    float* hi       = a_proj   + 512*320;  // 512*160
    float* hj       = hi       + 512*160;  // 512*160
    float* comb     = hj       + 512*160;  // 512*7
    float* r1       = comb     + 512*7;    // 512*128
    float* r2       = r1       + 512*128;  // 512*256

    const int WPB = 4;                     // waves per block
    const int TPB = WPB * 32;
    auto gemm = [&](const float* A, const float* W, const float* bias, float* C,
                    int M, int N, int K, int lda, bool relu) {
        const int tiles  = (M >> 5) * (N >> 5);   // 32x32 tiles per wave
        const int blocks = (tiles + WPB - 1) / WPB;
        if (relu)
            hipLaunchKernelGGL((wmma_gemm_f32<true>),  dim3(blocks), dim3(TPB), 0, stream,
                               A, W, bias, C, M, N, K, lda);
        else
            hipLaunchKernelGGL((wmma_gemm_f32<false>), dim3(blocks), dim3(TPB), 0, stream,
                               A, W, bias, C, M, N, K, lda);
    };

    // 1. backbone trunk: t1 = relu(x @ bb_w1 + bb_b1)   [512x640]
    gemm(x, bb_w1, bb_b1, t1, MROWS, 640, HDIM, HDIM, true);
    // 2. backbone = t1 @ bb_w2 + bb_b2                  [512x3]
    hipLaunchKernelGGL((small_gemm_f32<false>), dim3((MROWS*3 + 255)/256), dim3(256), 0, stream,
                       t1, bb_w2, bb_b2, bb_out, MROWS, 3, 640, 640);
    // 3. sc_in = concat(x, backbone)                    [512x323]
    hipLaunchKernelGGL(build_scin, dim3((MROWS*323 + 255)/256), dim3(256), 0, stream,
                       x, bb_out, scin);
    // 4. h1 = relu(sc_in @ sc_w1 + sc_b1)               [512x320], ragged K=323
    gemm(scin, sc_w1, sc_b1, h1, MROWS, HDIM, 323, 323, true);
    // 5. h2 = relu(h1 @ sc_w2 + sc_b2)                  [512x160]
    gemm(h1, sc_w2, sc_b2, h2, MROWS, 160, HDIM, HDIM, true);
    // 6. side_chain = h2 @ sc_w3 + sc_b3                [512x4]
    hipLaunchKernelGGL((small_gemm_f32<false>), dim3((MROWS*4 + 255)/256), dim3(256), 0, stream,
                       h2, sc_w3, sc_b3, sc_out, MROWS, 4, 160, 160);
    // 7. qkv = x @ in_proj_w + in_proj_b                [512x960]
    gemm(x, in_proj_w, in_proj_b, qkv, MROWS, 960, HDIM, HDIM, false);
    // 8. attention over S=2
    hipLaunchKernelGGL(attention2_kernel, dim3((SEQ_B*SEQ_L*NHEAD + 127)/128), dim3(128), 0, stream,
                       qkv, attn_out);
    // 9. a = attn_out @ out_proj_w + out_proj_b         [512x320]
    gemm(attn_out, out_proj_w, out_proj_b, a_proj, MROWS, HDIM, HDIM, HDIM, false);
    // 10/11. hi = a @ mlp_w1[:320], hj = a @ mlp_w1[320:]  (no bias)  [512x160]
    gemm(a_proj, mlp_w1,             nullptr, hi, MROWS, 160, HDIM, HDIM, false);
    gemm(a_proj, mlp_w1 + 320 * 160, nullptr, hj, MROWS, 160, HDIM, HDIM, false);
    // 12. contact map                                   [2x256x256]
    hipLaunchKernelGGL(contact_kernel, dim3(16, 16, 2), dim3(16, 16), 0, stream,
                       hi, hj, mlp_b1, mlp_w2, mlp_b2, cm_out);
    // 13. comb = concat(backbone, side_chain)           [512x7]
    hipLaunchKernelGGL(build_comb, dim3((MROWS*7 + 255)/256), dim3(256), 0, stream,
                       bb_out, sc_out, comb);
    // 14. r1 = relu(comb @ ref_w1 + ref_b1)             [512x128], ragged K=7
    gemm(comb, ref_w1, ref_b1, r1, MROWS, 128, 7, 7, true);
    // 15. r2 = relu(r1 @ ref_w2 + ref_b2)               [512x256]
    gemm(r1, ref_w2, ref_b2, r2, MROWS, 256, 128, 128, true);
    // 16. refined = r2 @ ref_w3 + ref_b3                [512x7]
    hipLaunchKernelGGL((small_gemm_f32<false>), dim3((MROWS*7 + 255)/256), dim3(256), 0, stream,
                       r2, ref_w3, ref_b3, rf_out, MROWS, 7, 256, 256);
}